// DGMC_modified_54314156425365
// MI455X (gfx1250) — compile-verified
//
#include <hip/hip_runtime.h>
#include <math.h>

// Problem constants (from reference)
#define Bsz 4
#define Nn 1024
#define C_IN 128
#define C_OUT 256
#define R_IN 32
#define R_OUT 32
#define NUM_PSI 2
#define Ee 65536
#define BN (Bsz * Nn)                 // 4096
#define SMAT ((size_t)Bsz * Nn * Nn)  // 4,194,304

typedef float v2f __attribute__((ext_vector_type(2)));
typedef float v8f __attribute__((ext_vector_type(8)));

#define WMMA_F32(a, b, c) \
  __builtin_amdgcn_wmma_f32_16x16x4_f32(false, (a), false, (b), (short)0, (c), false, false)

// ---------------- generic helpers ----------------

__global__ void copy_k(float* __restrict__ dst, const float* __restrict__ src, size_t n) {
  size_t i = (size_t)blockIdx.x * blockDim.x + threadIdx.x;
  if (i < n) dst[i] = src[i];
}

// tmp[dst[e]*C + f] += x[src[e]*C + f]
__global__ void scatter_add_k(float* __restrict__ tmp, const float* __restrict__ x,
                              const int* __restrict__ src, const int* __restrict__ dst,
                              int E, int C) {
  size_t tid = (size_t)blockIdx.x * blockDim.x + threadIdx.x;
  size_t total = (size_t)E * C;
  if (tid >= total) return;
  int e = (int)(tid / C);
  int f = (int)(tid % C);
  atomicAdd(&tmp[(size_t)dst[e] * C + f], x[(size_t)src[e] * C + f]);
}

// Row softmax fused with weighted accumulation.
//   v = softmax(in)[row]
//   if (soft_out) soft_out[row] = v
//   if (acc)      acc[row] = (init ? 0 : acc[row]) + w[widx]*v
// Safe for soft_out == in (in-place final softmax with acc == nullptr).
__global__ void softmax_acc_k(const float* __restrict__ in, float* __restrict__ soft_out,
                              float* __restrict__ acc, const float* __restrict__ w,
                              int widx, int init, int cols) {
  __shared__ float red[256];
  size_t row = blockIdx.x;
  const float* x = in + row * (size_t)cols;
  int tid = threadIdx.x;

  float m = -INFINITY;
  for (int c = tid; c < cols; c += 256) m = fmaxf(m, x[c]);
  red[tid] = m; __syncthreads();
  for (int s = 128; s > 0; s >>= 1) {
    if (tid < s) red[tid] = fmaxf(red[tid], red[tid + s]);
    __syncthreads();
  }
  m = red[0]; __syncthreads();

  float sum = 0.f;
  for (int c = tid; c < cols; c += 256) sum += __expf(x[c] - m);
  red[tid] = sum; __syncthreads();
  for (int s = 128; s > 0; s >>= 1) {
    if (tid < s) red[tid] += red[tid + s];
    __syncthreads();
  }
  float inv = 1.f / red[0];
  float wv = acc ? w[widx] : 0.f;

  for (int c = tid; c < cols; c += 256) {
    float v = __expf(x[c] - m) * inv;
    size_t idx = row * (size_t)cols + c;
    if (soft_out) soft_out[idx] = v;
    if (acc) acc[idx] = (init ? 0.f : acc[idx]) + wv * v;
  }
}

// ---------------- WMMA GEMMs (f32 16x16x4, 32x64 tile per wave) ----------------

// C[M,N] = relu(A[M,K] @ W[K,N] + bias[N]); 32(M) x 64(N) tile per wave32.
__global__ void gemm_bias_relu_wmma(const float* __restrict__ A, const float* __restrict__ W,
                                    const float* __restrict__ bias, float* __restrict__ C,
                                    int M, int K, int N) {
  int wave = blockIdx.x * (blockDim.x >> 5) + (threadIdx.x >> 5);
  int lane = threadIdx.x & 31;
  int ntN = N >> 6;                       // tiles of 64 in N
  int tm = wave / ntN, tn = wave % ntN;
  if (tm * 32 >= M) return;

  int r16 = lane & 15;                    // A row / B col / C col (within 16)
  int kb  = (lane >> 4) << 1;             // k sub-offset {0,2}
  const float* Ar0 = A + (size_t)(tm * 32 + r16) * K;
  const float* Ar1 = Ar0 + (size_t)16 * K;

  v8f acc[2][4] = {};
  for (int k0 = 0; k0 < K; k0 += 4) {
    v2f a0, a1, bf[4];
    a0.x = Ar0[k0 + kb]; a0.y = Ar0[k0 + kb + 1];
    a1.x = Ar1[k0 + kb]; a1.y = Ar1[k0 + kb + 1];
    const float* Wr0 = W + (size_t)(k0 + kb) * N + tn * 64 + r16;
    const float* Wr1 = Wr0 + N;
#pragma unroll
    for (int n = 0; n < 4; ++n) { bf[n].x = Wr0[n * 16]; bf[n].y = Wr1[n * 16]; }
#pragma unroll
    for (int n = 0; n < 4; ++n) {
      acc[0][n] = WMMA_F32(a0, bf[n], acc[0][n]);
      acc[1][n] = WMMA_F32(a1, bf[n], acc[1][n]);
    }
  }
  int rbase = (lane >> 4) * 8;
#pragma unroll
  for (int m = 0; m < 2; ++m)
#pragma unroll
    for (int n = 0; n < 4; ++n) {
      float bv = bias[tn * 64 + n * 16 + r16];
#pragma unroll
      for (int v = 0; v < 8; ++v) {
        float xv = acc[m][n][v] + bv;
        C[(size_t)(tm * 32 + m * 16 + rbase + v) * N + tn * 64 + n * 16 + r16] =
            fmaxf(xv, 0.f);
      }
    }
}

// S[b] = hs[b] @ ht[b]^T ; hs,ht: [B,1024,256]; S: [B,1024,1024].
// 32(s) x 64(t) tile per wave32; B fragment reads ht rows (contiguous in k).
__global__ void shat_wmma(const float* __restrict__ hs, const float* __restrict__ ht,
                          float* __restrict__ S) {
  int wave = blockIdx.x * (blockDim.x >> 5) + (threadIdx.x >> 5);
  int lane = threadIdx.x & 31;
  int b  = wave >> 9;                     // (1024/32)*(1024/64)=512 tiles per batch
  int tm = (wave >> 4) & 31;
  int tn = wave & 15;

  int r16 = lane & 15;
  int kb  = (lane >> 4) << 1;
  const float* A0 = hs + ((size_t)b * Nn + tm * 32 + r16) * C_OUT;
  const float* A1 = A0 + (size_t)16 * C_OUT;
  const float* Bp[4];
#pragma unroll
  for (int n = 0; n < 4; ++n)
    Bp[n] = ht + ((size_t)b * Nn + tn * 64 + n * 16 + r16) * C_OUT;

  v8f acc[2][4] = {};
  for (int k0 = 0; k0 < C_OUT; k0 += 4) {
    v2f a0, a1, bf[4];
    a0.x = A0[k0 + kb]; a0.y = A0[k0 + kb + 1];
    a1.x = A1[k0 + kb]; a1.y = A1[k0 + kb + 1];
#pragma unroll
    for (int n = 0; n < 4; ++n) { bf[n].x = Bp[n][k0 + kb]; bf[n].y = Bp[n][k0 + kb + 1]; }
#pragma unroll
    for (int n = 0; n < 4; ++n) {
      acc[0][n] = WMMA_F32(a0, bf[n], acc[0][n]);
      acc[1][n] = WMMA_F32(a1, bf[n], acc[1][n]);
    }
  }
  int rbase = (lane >> 4) * 8;
  float* Sout = S + (size_t)b * Nn * Nn;
#pragma unroll
  for (int m = 0; m < 2; ++m)
#pragma unroll
    for (int n = 0; n < 4; ++n)
#pragma unroll
      for (int v = 0; v < 8; ++v)
        Sout[(size_t)(tm * 32 + m * 16 + rbase + v) * Nn + tn * 64 + n * 16 + r16] =
            acc[m][n][v];
}

// ---------------- small / fused kernels ----------------

// r_t[b,t,j] = sum_s soft[b,s,t] * r_s[b,s,j]; 256 thr = 8 t-rows x 32 j.
__global__ void rt_k(const float* __restrict__ soft, const float* __restrict__ rs,
                     float* __restrict__ rt) {
  int idx = blockIdx.x * 8 + (threadIdx.x >> 5); // global b*1024+t
  int b = idx >> 10, t = idx & 1023;
  int j = threadIdx.x & 31;
  const float* S = soft + (size_t)b * Nn * Nn;
  const float* R = rs + (size_t)b * Nn * R_IN;
  float acc = 0.f;
  for (int s = 0; s < Nn; ++s) acc += S[(size_t)s * Nn + t] * R[s * R_IN + j];
  rt[((size_t)b * Nn + t) * R_IN + j] = acc;
}

// P[row] = relu(tmp[row] @ Wp + bp) @ Wm1 ; 128 thr = 4 rows x 32 cols.
__global__ void psi_small_k(const float* __restrict__ tmp, const float* __restrict__ Wp,
                            const float* __restrict__ bp, const float* __restrict__ Wm1,
                            float* __restrict__ P) {
  __shared__ float o[4][R_OUT];
  int r = threadIdx.x >> 5, j = threadIdx.x & 31;
  int row = blockIdx.x * 4 + r;
  const float* t = tmp + (size_t)row * R_IN;
  float acc = bp[j];
  for (int k = 0; k < R_IN; ++k) acc += t[k] * Wp[k * R_OUT + j];
  o[r][j] = fmaxf(acc, 0.f);
  __syncthreads();
  float p = 0.f;
  for (int k = 0; k < R_OUT; ++k) p += o[r][k] * Wm1[k * R_OUT + j];
  P[(size_t)row * R_OUT + j] = p;
}

// S_hat[b,s,t] += bm2 + sum_k relu(Ps[b,s,k] - Pt[b,t,k] + bm1[k]) * Wm2[k]
// 16x16 (s,t) tile per block of 256.
__global__ void pair_upd_k(const float* __restrict__ Ps, const float* __restrict__ Pt,
                           const float* __restrict__ bm1, const float* __restrict__ Wm2,
                           const float* __restrict__ bm2, float* __restrict__ Shat) {
  __shared__ float ps[16][R_OUT], pt[16][R_OUT], sb[R_OUT], wb[R_OUT];
  int blk = blockIdx.x;
  int b = blk >> 12, ts = (blk >> 6) & 63, tt = blk & 63;
  int tid = threadIdx.x;
  for (int i = tid; i < 16 * R_OUT; i += 256) {
    int rr = i >> 5, cc = i & 31;
    ps[rr][cc] = Ps[((size_t)b * Nn + ts * 16 + rr) * R_OUT + cc];
    pt[rr][cc] = Pt[((size_t)b * Nn + tt * 16 + rr) * R_OUT + cc];
  }
  if (tid < R_OUT) { sb[tid] = bm1[tid]; wb[tid] = Wm2[tid]; }
  __syncthreads();
  int si = tid >> 4, ti = tid & 15;
  float u = bm2[0];
#pragma unroll
  for (int k = 0; k < R_OUT; ++k)
    u += fmaxf(ps[si][k] - pt[ti][k] + sb[k], 0.f) * wb[k];
  Shat[((size_t)b * Nn + ts * 16 + si) * Nn + tt * 16 + ti] += u;
}

// ---------------- driver ----------------

extern "C" void kernel_launch(void* const* d_in, const int* in_sizes, int n_in,
                              void* d_out, int out_size, void* d_ws, size_t ws_size,
                              hipStream_t stream) {
  const float* x_s  = (const float*)d_in[0];
  const int*   ei_s = (const int*)d_in[1];
  const float* x_t  = (const float*)d_in[2];
  const int*   ei_t = (const int*)d_in[3];
  const float* r    = (const float*)d_in[4];   // [2,B,N,32]
  const float* W1   = (const float*)d_in[5];
  const float* b1   = (const float*)d_in[6];
  const float* Wp   = (const float*)d_in[7];   // [2,32,32]
  const float* bp   = (const float*)d_in[8];   // [2,32]
  const float* Wm1  = (const float*)d_in[9];
  const float* bm1  = (const float*)d_in[10];
  const float* Wm2  = (const float*)d_in[11];
  const float* bm2  = (const float*)d_in[12];
  const float* sw   = (const float*)d_in[13];

  float* ws   = (float*)d_ws;
  float* tmp  = ws;                         // 524288 (reused: C_IN psi + R_IN psi)
  float* hs   = tmp + (size_t)BN * C_IN;    // 1048576
  float* ht   = hs + (size_t)BN * C_OUT;    // 1048576
  float* Shat = ht + (size_t)BN * C_OUT;    // 4194304
  float* soft = Shat + SMAT;                // 4194304
  float* rt   = soft + SMAT;                // 131072
  float* Ps   = rt + (size_t)BN * R_IN;     // 131072
  float* Pt   = Ps + (size_t)BN * R_OUT;    // 131072
  float* outF = (float*)d_out;              // doubles as S_final accumulator

  const int* src_s = ei_s;           const int* dst_s = ei_s + Ee;
  const int* src_t = ei_t;           const int* dst_t = ei_t + Ee;

  size_t nBig = (size_t)BN * C_IN;
  size_t nSm  = (size_t)BN * R_IN;

  // waves: (M/32)*(N/64); 8 waves per 256-thread block
  const int gemm1_blocks = (BN / 32) * (C_OUT / 64) / 8;       // 64
  const int shat_blocks  = Bsz * (Nn / 32) * (Nn / 64) / 8;    // 256

  // ---- psi1 on source graph -> hs
  copy_k<<<(int)((nBig + 255) / 256), 256, 0, stream>>>(tmp, x_s, nBig);
  scatter_add_k<<<(int)(((size_t)Ee * C_IN + 255) / 256), 256, 0, stream>>>(tmp, x_s, src_s, dst_s, Ee, C_IN);
  gemm_bias_relu_wmma<<<gemm1_blocks, 256, 0, stream>>>(tmp, W1, b1, hs, BN, C_IN, C_OUT);
  // ---- psi1 on target graph -> ht
  copy_k<<<(int)((nBig + 255) / 256), 256, 0, stream>>>(tmp, x_t, nBig);
  scatter_add_k<<<(int)(((size_t)Ee * C_IN + 255) / 256), 256, 0, stream>>>(tmp, x_t, src_t, dst_t, Ee, C_IN);
  gemm_bias_relu_wmma<<<gemm1_blocks, 256, 0, stream>>>(tmp, W1, b1, ht, BN, C_IN, C_OUT);

  // ---- S_hat = hs @ ht^T (batched)
  shat_wmma<<<shat_blocks, 256, 0, stream>>>(hs, ht, Shat);

  // ---- S_0: soft = softmax(S_hat), out = w0*soft (fused)
  softmax_acc_k<<<BN, 256, 0, stream>>>(Shat, soft, outF, sw, 0, 1, Nn);

  for (int i = 0; i < NUM_PSI; ++i) {
    const float* r_i  = r + (size_t)i * BN * R_IN;
    const float* Wp_i = Wp + (size_t)i * R_IN * R_OUT;
    const float* bp_i = bp + (size_t)i * R_OUT;

    // r_t = soft^T @ r_s (per batch); soft currently == softmax(S_hat)
    rt_k<<<BN / 8, 256, 0, stream>>>(soft, r_i, rt);

    // P_s = (relu((r_i + agg_s) @ Wp + bp)) @ Wm1
    copy_k<<<(int)((nSm + 255) / 256), 256, 0, stream>>>(tmp, r_i, nSm);
    scatter_add_k<<<(int)(((size_t)Ee * R_IN + 255) / 256), 256, 0, stream>>>(tmp, r_i, src_s, dst_s, Ee, R_IN);
    psi_small_k<<<BN / 4, 128, 0, stream>>>(tmp, Wp_i, bp_i, Wm1, Ps);

    // P_t = (relu((r_t + agg_t) @ Wp + bp)) @ Wm1
    copy_k<<<(int)((nSm + 255) / 256), 256, 0, stream>>>(tmp, rt, nSm);
    scatter_add_k<<<(int)(((size_t)Ee * R_IN + 255) / 256), 256, 0, stream>>>(tmp, rt, src_t, dst_t, Ee, R_IN);
    psi_small_k<<<BN / 4, 128, 0, stream>>>(tmp, Wp_i, bp_i, Wm1, Pt);

    // S_hat += pairwise MLP (factored through Wm1)
    pair_upd_k<<<Bsz * 64 * 64, 256, 0, stream>>>(Ps, Pt, bm1, Wm2, bm2, Shat);

    // soft = softmax(S_hat) (skipped on last iter: nobody consumes it),
    // out += w[i+1]*softmax(S_hat)   (fused single pass)
    float* soft_dst = (i == NUM_PSI - 1) ? nullptr : soft;
    softmax_acc_k<<<BN, 256, 0, stream>>>(Shat, soft_dst, outF, sw, i + 1, 0, Nn);
  }

  // final row softmax over [B*N, N], in place on d_out (acc == nullptr)
  softmax_acc_k<<<BN, 256, 0, stream>>>(outF, outF, nullptr, sw, 0, 0, Nn);
}